// DecoupleModel_20873541059014
// MI455X (gfx1250) — compile-verified
//
#include <hip/hip_runtime.h>

// ---------------------------------------------------------------------------
// DecoupleModel on MI455X (gfx1250).
// GEMMs: V_WMMA_F32_16X16X4_F32 (full fp32 on the matrix pipe).
// X tiles staged with GLOBAL_LOAD_ASYNC_TO_LDS_B128 (ASYNCcnt-tracked).
// W tiles staged k-pair-interleaved so B fragments are single ds_load_b64.
// Each wave register-blocks 32x64 (2 A frags x 4 B frags -> 8 WMMA / k-step).
// Edge aggregation: b128 gather + hardware global_atomic_add_f32.
// ---------------------------------------------------------------------------

typedef float v2f __attribute__((ext_vector_type(2)));
typedef float v8f __attribute__((ext_vector_type(8)));
typedef int   v4i __attribute__((vector_size(16)));   // matches builtin proto

#define N_NODES 10000
#define NE      320000
#define DIM_IN  256
#define DIM_MP  256
#define DIM_FL  512
#define DIM_OUT 64

#if defined(__HIP_DEVICE_COMPILE__) && __has_builtin(__builtin_amdgcn_global_load_async_to_lds_b128)
#define USE_ASYNC_LDS 1
#else
#define USE_ASYNC_LDS 0
#endif

#if USE_ASYNC_LDS
__device__ __forceinline__ void async_copy_b128(const void* g, void* l) {
  // per-lane copy of 16 bytes from global address g to LDS address l, tracked by ASYNCcnt
  __builtin_amdgcn_global_load_async_to_lds_b128(
      (__attribute__((address_space(1))) v4i*)(g),
      (__attribute__((address_space(3))) v4i*)(l), 0, 0);
}
__device__ __forceinline__ void wait_async() {
#if __has_builtin(__builtin_amdgcn_s_wait_asynccnt)
  __builtin_amdgcn_s_wait_asynccnt(0);
#else
  asm volatile("s_wait_asynccnt 0x0" ::: "memory");
#endif
}
#endif

// ---------------------------------------------------------------------------
// Generic WMMA GEMM:
//   Y[row, 0..M) (=/+=) rowScale[row] * alpha * post( act(X[row,0..K)) @ W + bias )
// Block = 128 threads = 4 waves. Tile: 128 (M) x 64 (N), BK = 32.
// Wave w computes rows [w*32, w*32+32) x all 64 cols (2x4 v8f accumulators).
// Only Nrows needs bounds guarding; K and M are multiples of 32/64.
// ---------------------------------------------------------------------------
template<bool RELU_IN, bool RELU_OUT, bool ACCUM>
__global__ __launch_bounds__(128) void gemm_wmma(
    const float* __restrict__ X, const float* __restrict__ W,
    const float* __restrict__ bias, const float* __restrict__ rowScale,
    const float* __restrict__ alphaPtr, float* __restrict__ Y,
    int Nrows, int K, int M)
{
  // Xs stride 36: 36*r mod 64 distinct over any 16 consecutive rows ->
  // conflict-free A reads; rows stay 16B-aligned (144B row pitch).
  __shared__ float  Xs[128][36];
  // Wp[p][c] = { W[2p][c], W[2p+1][c] }; row stride 80 float2 = 160 dwords
  // == 32 mod 64 banks -> the two half-waves use disjoint bank halves.
  __shared__ float2 Wp[16][80];

  const int tid  = threadIdx.x;
  const int lane = tid & 31;
  const int wave = tid >> 5;
  const int blockRow = blockIdx.y * 128;
  const int blockCol = blockIdx.x * 64;

  const v8f vzero = {0.f,0.f,0.f,0.f,0.f,0.f,0.f,0.f};
  v8f acc[2][4] = {{vzero, vzero, vzero, vzero}, {vzero, vzero, vzero, vzero}};

  // X staging: one tile row per thread, 8 x 16B each
  const int  gRow  = blockRow + tid;
  const bool rowOK = (gRow < Nrows);
  const int  gRowC = rowOK ? gRow : (Nrows - 1);   // clamped (safe, unused rows)

  // W staging: thread -> (k-pair p, 8-col group), 4 x float4 -> 8 x float2
  const int wp = tid >> 3;             // 0..15
  const int wg = (tid & 7) * 8;        // 0,8,...,56

  // per-lane WMMA fragment coordinates (ISA 16x16x4 f32 layout)
  const int arow0 = wave * 32 + (lane & 15);   // first 16-row group
  const int koff  = (lane >> 4) * 2;           // lanes 16..31 hold K+2, K+3
  const int bcol  = lane & 15;

  for (int k0 = 0; k0 < K; k0 += 32) {
    __syncthreads();

    // ---- stage X tile (128x32) ----
#if USE_ASYNC_LDS
    if (!RELU_IN) {
#pragma unroll
      for (int i = 0; i < 8; ++i) {
        async_copy_b128(X + (size_t)gRowC * K + (k0 + i * 4), &Xs[tid][i * 4]);
      }
    } else
#endif
    {
#pragma unroll
      for (int i = 0; i < 8; ++i) {
        float4 v = make_float4(0.f, 0.f, 0.f, 0.f);
        if (rowOK)
          v = *(const float4*)(X + (size_t)gRow * K + (k0 + i * 4));
        if (RELU_IN) {
          v.x = fmaxf(v.x, 0.f); v.y = fmaxf(v.y, 0.f);
          v.z = fmaxf(v.z, 0.f); v.w = fmaxf(v.w, 0.f);
        }
        Xs[tid][i * 4 + 0] = v.x; Xs[tid][i * 4 + 1] = v.y;
        Xs[tid][i * 4 + 2] = v.z; Xs[tid][i * 4 + 3] = v.w;
      }
    }

    // ---- stage W tile (32x64) k-pair interleaved ----
    {
      const float* w0 = W + (size_t)(k0 + 2 * wp    ) * M + blockCol + wg;
      const float* w1 = W + (size_t)(k0 + 2 * wp + 1) * M + blockCol + wg;
      float4 a0 = *(const float4*)(w0);
      float4 a1 = *(const float4*)(w0 + 4);
      float4 b0 = *(const float4*)(w1);
      float4 b1 = *(const float4*)(w1 + 4);
      Wp[wp][wg + 0] = make_float2(a0.x, b0.x);
      Wp[wp][wg + 1] = make_float2(a0.y, b0.y);
      Wp[wp][wg + 2] = make_float2(a0.z, b0.z);
      Wp[wp][wg + 3] = make_float2(a0.w, b0.w);
      Wp[wp][wg + 4] = make_float2(a1.x, b1.x);
      Wp[wp][wg + 5] = make_float2(a1.y, b1.y);
      Wp[wp][wg + 6] = make_float2(a1.z, b1.z);
      Wp[wp][wg + 7] = make_float2(a1.w, b1.w);
    }

#if USE_ASYNC_LDS
    if (!RELU_IN) wait_async();
#endif
    __syncthreads();

    // ---- 8 k-steps; per step: 2 A frags x 4 B frags -> 8 WMMA ----
#pragma unroll
    for (int kk = 0; kk < 32; kk += 4) {
      v2f a0, a1;
      a0.x = Xs[arow0     ][kk + koff];
      a0.y = Xs[arow0     ][kk + koff + 1];
      a1.x = Xs[arow0 + 16][kk + koff];
      a1.y = Xs[arow0 + 16][kk + koff + 1];
      const int kp = (kk + koff) >> 1;
#pragma unroll
      for (int nt = 0; nt < 4; ++nt) {
        const float2 bw = Wp[kp][nt * 16 + bcol];
        v2f b; b.x = bw.x; b.y = bw.y;
        acc[0][nt] = __builtin_amdgcn_wmma_f32_16x16x4_f32(
            false, a0, false, b, (short)0, acc[0][nt], false, false);
        acc[1][nt] = __builtin_amdgcn_wmma_f32_16x16x4_f32(
            false, a1, false, b, (short)0, acc[1][nt], false, false);
      }
    }
  }

  // ---- epilogue ----
  const float alphaV = alphaPtr ? alphaPtr[0] : 1.0f;
  const int   mOff   = (lane >> 4) * 8;   // lanes 16..31 hold M = j+8
#pragma unroll
  for (int mb = 0; mb < 2; ++mb) {
#pragma unroll
    for (int nt = 0; nt < 4; ++nt) {
      const int col = blockCol + nt * 16 + (lane & 15);
      const float bv = bias ? bias[col] : 0.0f;
#pragma unroll
      for (int j = 0; j < 8; ++j) {
        const int row = blockRow + wave * 32 + mb * 16 + mOff + j;
        if (row < Nrows) {
          float v = acc[mb][nt][j] + bv;
          if (RELU_OUT) v = fmaxf(v, 0.0f);
          v *= alphaV;
          if (rowScale) v *= rowScale[row];
          float* p = Y + (size_t)row * M + col;
          if (ACCUM) *p += v; else *p = v;
        }
      }
    }
  }
}

// ---------------------------------------------------------------------------
// Degree histogram over col, then invert with clamp.
// ---------------------------------------------------------------------------
__global__ void deg_count_kernel(const int* __restrict__ col,
                                 float* __restrict__ deg, int E) {
  int i = blockIdx.x * blockDim.x + threadIdx.x;
  if (i < E) unsafeAtomicAdd(&deg[col[i]], 1.0f);
}

__global__ void deg_inv_kernel(float* __restrict__ deg, int n) {
  int i = blockIdx.x * blockDim.x + threadIdx.x;
  if (i < n) deg[i] = 1.0f / fmaxf(deg[i], 1.0f);
}

// ---------------------------------------------------------------------------
// out[row[e], :] += H[col[e], :]
// 4 edges per 256-thread block; 64 lanes/edge, float4 gather + 4 fp32 atomics.
// ---------------------------------------------------------------------------
__global__ __launch_bounds__(256) void scatter_add_kernel(
    const float* __restrict__ H, const int* __restrict__ row,
    const int* __restrict__ col, float* __restrict__ out) {
  const int e = blockIdx.x * 4 + (threadIdx.x >> 6);
  const int q = (threadIdx.x & 63) * 4;
  const int r = row[e];
  const int c = col[e];
  const float4 v = *(const float4*)(H + (size_t)c * DIM_MP + q);
  float* o = out + (size_t)r * DIM_MP + q;
  unsafeAtomicAdd(o + 0, v.x);
  unsafeAtomicAdd(o + 1, v.y);
  unsafeAtomicAdd(o + 2, v.z);
  unsafeAtomicAdd(o + 3, v.w);
}

// ---------------------------------------------------------------------------
extern "C" void kernel_launch(void* const* d_in, const int* in_sizes, int n_in,
                              void* d_out, int out_size, void* d_ws, size_t ws_size,
                              hipStream_t stream) {
  (void)in_sizes; (void)n_in; (void)out_size; (void)ws_size;

  const float* x      = (const float*)d_in[0];
  const int*   ei     = (const int*)d_in[1];   // [2, E]: row = ei[0:E], col = ei[E:2E]
  const float* mp_w0  = (const float*)d_in[2];
  const float* mp_b0  = (const float*)d_in[3];
  const float* mp_w1  = (const float*)d_in[4];
  const float* mp_b1  = (const float*)d_in[5];
  const float* mp_w2  = (const float*)d_in[6];
  const float* mp_b2  = (const float*)d_in[7];
  const float* fc_w0  = (const float*)d_in[8];
  const float* fc_b0  = (const float*)d_in[9];
  const float* fc_w1  = (const float*)d_in[10];
  const float* fc_b1  = (const float*)d_in[11];
  const float* inj_w0 = (const float*)d_in[12];
  const float* inj_b0 = (const float*)d_in[13];
  const float* inj_w1 = (const float*)d_in[14];
  const float* inj_b1 = (const float*)d_in[15];
  const float* alpha  = (const float*)d_in[16];
  const float* out_w  = (const float*)d_in[17];
  const float* out_b  = (const float*)d_in[18];

  // workspace layout (floats): deg | P(N*MP) | Q(N*MP) | C(N*FL) | D(N*FL)
  float* ws  = (float*)d_ws;
  float* deg = ws;
  float* P   = ws + N_NODES;
  float* Q   = P + (size_t)N_NODES * DIM_MP;
  float* C   = Q + (size_t)N_NODES * DIM_MP;
  float* D   = C + (size_t)N_NODES * DIM_FL;

  const int* erow = ei;
  const int* ecol = ei + NE;

  // degree -> deg_inv
  (void)hipMemsetAsync(deg, 0, N_NODES * sizeof(float), stream);
  deg_count_kernel<<<(NE + 255) / 256, 256, 0, stream>>>(ecol, deg, NE);
  deg_inv_kernel<<<(N_NODES + 255) / 256, 256, 0, stream>>>(deg, N_NODES);

  const dim3 blk(128);
  const dim3 gMP (DIM_MP  / 64, (N_NODES + 127) / 128);
  const dim3 gFL (DIM_FL  / 64, (N_NODES + 127) / 128);
  const dim3 gOUT(DIM_OUT / 64, (N_NODES + 127) / 128);
  const size_t mpBytes = (size_t)N_NODES * DIM_MP * sizeof(float);

  // ---- MP layer 0: Q = relu(x@w0 + b0) * deg_inv ; P = scatter(Q) ----
  gemm_wmma<false, true, false><<<gMP, blk, 0, stream>>>(
      x, mp_w0, mp_b0, deg, nullptr, Q, N_NODES, DIM_IN, DIM_MP);
  (void)hipMemsetAsync(P, 0, mpBytes, stream);
  scatter_add_kernel<<<NE / 4, 256, 0, stream>>>(Q, erow, ecol, P);

  // ---- MP layer 1 ----
  gemm_wmma<false, true, false><<<gMP, blk, 0, stream>>>(
      P, mp_w1, mp_b1, deg, nullptr, Q, N_NODES, DIM_MP, DIM_MP);
  (void)hipMemsetAsync(P, 0, mpBytes, stream);
  scatter_add_kernel<<<NE / 4, 256, 0, stream>>>(Q, erow, ecol, P);

  // ---- MP layer 2 (result h3 lives in P; used 3x by the FC stage) ----
  gemm_wmma<false, true, false><<<gMP, blk, 0, stream>>>(
      P, mp_w2, mp_b2, deg, nullptr, Q, N_NODES, DIM_MP, DIM_MP);
  (void)hipMemsetAsync(P, 0, mpBytes, stream);
  scatter_add_kernel<<<NE / 4, 256, 0, stream>>>(Q, erow, ecol, P);

  // ---- FC1: C = alpha*(relu(h3)@fc_w0 + fc_b0) + (h3@inj_w0 + inj_b0) ----
  gemm_wmma<true, false, false><<<gFL, blk, 0, stream>>>(
      P, fc_w0, fc_b0, nullptr, alpha, C, N_NODES, DIM_MP, DIM_FL);
  gemm_wmma<false, false, true><<<gFL, blk, 0, stream>>>(
      P, inj_w0, inj_b0, nullptr, nullptr, C, N_NODES, DIM_MP, DIM_FL);

  // ---- FC2: D = alpha*(relu(C)@fc_w1 + fc_b1) + (h3@inj_w1 + inj_b1) ----
  gemm_wmma<true, false, false><<<gFL, blk, 0, stream>>>(
      C, fc_w1, fc_b1, nullptr, alpha, D, N_NODES, DIM_FL, DIM_FL);
  gemm_wmma<false, false, true><<<gFL, blk, 0, stream>>>(
      P, inj_w1, inj_b1, nullptr, nullptr, D, N_NODES, DIM_MP, DIM_FL);

  // ---- out: d_out = D @ out_w + out_b ----
  gemm_wmma<false, false, false><<<gOUT, blk, 0, stream>>>(
      D, out_w, out_b, nullptr, nullptr, (float*)d_out, N_NODES, DIM_FL, DIM_OUT);
}